// LstmGcnNet_54477365183266
// MI455X (gfx1250) — compile-verified
//
#include <hip/hip_runtime.h>
#include <math.h>

// ---------------- problem constants (from reference) ----------------
#define S_T   12
#define N_N   16384
#define E_E   262144
#define D_D   128          // DIN = DOUT = H
#define B_B   64
#define T_T   3072         // S*N/B
#define ROWS  196608       // S*N

typedef __attribute__((ext_vector_type(2))) float        v2f;
typedef __attribute__((ext_vector_type(8))) float        v8f;
typedef __attribute__((ext_vector_type(4))) int          v4i;
typedef __attribute__((ext_vector_type(8))) int          v8i;
typedef __attribute__((ext_vector_type(4))) unsigned int v4u;

// pointer flavors for the async-to-LDS builtin (probed prototype: v4i* args)
typedef __attribute__((address_space(1))) v4i* gl_v4i_p;   // global
typedef __attribute__((address_space(3))) v4i* ld_v4i_p;   // LDS

__device__ __forceinline__ float sigmoidf_(float x) {
  return 1.0f / (1.0f + __expf(-x));
}

// native fp32 atomic add (GLOBAL_ATOMIC_ADD_F32), not a CAS loop
__device__ __forceinline__ void atom_add_f32(float* p, float v) {
  unsafeAtomicAdd(p, v);
}

// wait for all outstanding async (global<->LDS) transfers
__device__ __forceinline__ void wait_async0() {
#if __has_builtin(__builtin_amdgcn_s_wait_asynccnt)
  __builtin_amdgcn_s_wait_asynccnt(0);
#else
  asm volatile("s_wait_asynccnt 0x0" ::: "memory");
#endif
}

// ---------------- zero d_out (needed: scatter accumulates into it) ----------
__global__ void zero_f4(float4* __restrict__ p, int n4) {
  int i = blockIdx.x * blockDim.x + threadIdx.x;
  int stride = gridDim.x * blockDim.x;
  float4 z; z.x = z.y = z.z = z.w = 0.0f;
  for (; i < n4; i += stride) p[i] = z;
}

// ---------------- GCN dense GEMM: support = xs @ W  (fp32 WMMA 16x16x4) ----
// One block = 8 waves; each wave owns one 16x16 output tile (16 rows x full
// 128 cols split across 8 waves). A tile staged in LDS two ways:
//   rows 0..7  : Tensor Data Mover (tensor_load_to_lds, TENSORcnt), with
//                descriptor padding 128dw+4dw reproducing the [16][132] stride
//   rows 8..15 : global_load_async_to_lds_b128 (ASYNCcnt)
// W read from L2 (64 KB, fully resident).
__global__ __launch_bounds__(256) void support_gemm(
    const float* __restrict__ xs, const float* __restrict__ W,
    float* __restrict__ sup) {
  // pad 132: rows 528 B (16B multiple for B128), 132%64==4 -> fragment reads
  // at[l16][kidx] hit banks (4*l16 + kidx)%64: fully conflict-free.
  __shared__ __align__(16) float at[16][132];
  const int tid  = threadIdx.x;
  const int lane = tid & 31;
  const int wv   = tid >> 5;
  const int hf   = lane >> 4;     // which 16-lane half
  const int l16  = lane & 15;
  const int rowbase = blockIdx.x * 16;

#if __has_builtin(__builtin_amdgcn_tensor_load_to_lds) && \
    __has_builtin(__builtin_amdgcn_s_wait_tensorcnt) &&    \
    __has_builtin(__builtin_amdgcn_global_load_async_to_lds_b128)
  if (tid == 0) {
    // ---- D# for an 8x128 f32 tile, LDS-padded to stride 132 dwords ----
    unsigned lds_base = (unsigned)(size_t)&at[0][0];   // low 32b = LDS offset
    unsigned long long ga =
        (unsigned long long)(const void*)(xs + (size_t)rowbase * D_D);
    v4u g0;
    g0.x = 1u;                                   // count=1, user descriptor
    g0.y = lds_base;                             // lds_addr
    g0.z = (unsigned)(ga & 0xffffffffu);         // global_addr[31:0]
    g0.w = (unsigned)((ga >> 32) & 0x01ffffffu)  // global_addr[56:32]
         | (2u << 30);                           // type = 2 ("image")
    v8i g1;
    g1[0] = (int)((2u << 16)     // data_size = 4 B
                | (1u << 20)     // pad_enable
                | (6u << 22)     // pad_interval: 2^(6+1)=128 dwords (one row)
                | (3u << 25));   // pad_amount : 3+1 = 4 dwords -> stride 132
    g1[1] = (int)(128u << 16);   // tensor_dim0[15:0]=128 (bits 79:64... lo16)
    g1[2] = 0;                   // tensor_dim0 hi16 | tensor_dim1 lo16 (=0)
    g1[3] = (int)((128u << 16)   // tile_dim0 = 128
                | 0x3u);         // tensor_dim1 hi16 -> dim1 = 0x30000 = ROWS
    g1[4] = 8;                   // tile_dim1 = 8 rows (TDM half of the tile)
    g1[5] = 128;                 // tensor_dim0_stride lo32 = 128 elements
    g1[6] = 0;
    g1[7] = 0;
    v4i z4 = {0, 0, 0, 0};
#if __clang_major__ >= 23
    v8i z8 = {0, 0, 0, 0, 0, 0, 0, 0};
    __builtin_amdgcn_tensor_load_to_lds(g0, g1, z4, z4, z8, 0);
#else
    __builtin_amdgcn_tensor_load_to_lds(g0, g1, z4, z4, 0);
#endif
    __builtin_amdgcn_s_wait_tensorcnt(0);
  }
  // ---- rows 8..15 via async-to-LDS B128 (256 transfers of 16 B) ----
  for (int i = 256 + tid; i < 512; i += 256) {
    int r  = i >> 5;              // row 8..15
    int c4 = (i & 31) * 4;        // float col 0,4,...,124
    __builtin_amdgcn_global_load_async_to_lds_b128(
        (gl_v4i_p)(xs + (rowbase + r) * D_D + c4),
        (ld_v4i_p)&at[r][c4], 0, 0);
  }
  wait_async0();
  __syncthreads();
#elif __has_builtin(__builtin_amdgcn_global_load_async_to_lds_b128)
  for (int i = tid; i < 512; i += 256) {
    int r  = i >> 5;
    int c4 = (i & 31) * 4;
    __builtin_amdgcn_global_load_async_to_lds_b128(
        (gl_v4i_p)(xs + (rowbase + r) * D_D + c4),
        (ld_v4i_p)&at[r][c4], 0, 0);
  }
  wait_async0();
  __syncthreads();
#else
  for (int i = tid; i < 16 * 64; i += 256) {   // float2 granules, coalesced
    int r  = i >> 6;
    int c2 = (i & 63) * 2;
    const float* g = xs + (rowbase + r) * D_D + c2;
    at[r][c2]     = g[0];
    at[r][c2 + 1] = g[1];
  }
  __syncthreads();
#endif

  const int n = wv * 16 + l16;    // output column for this lane
  v8f acc = {};
  for (int kk = 0; kk < 32; ++kk) {
    int kidx = kk * 4 + hf * 2;   // k-pair this lane supplies
    v2f a;
    a.x = at[l16][kidx];
    a.y = at[l16][kidx + 1];
    v2f b;
    b.x = W[kidx * D_D + n];
    b.y = W[(kidx + 1) * D_D + n];
    acc = __builtin_amdgcn_wmma_f32_16x16x4_f32(false, a, false, b,
                                                (short)0, acc, false, false);
  }
  // store D tile: reg r -> M = r + 8*half, N = n
  float* o = sup + rowbase * D_D;
  #pragma unroll
  for (int r = 0; r < 8; ++r) {
    int m = r + hf * 8;
    o[m * D_D + n] = acc[r];
  }
}

// ---------------- GCN sparse scatter: out[idx0] += val * support[idx1] ------
// One wave per edge; each lane handles 4 of the 128 channels (float4 gather),
// accumulation via native GLOBAL_ATOMIC_ADD_F32.
__global__ __launch_bounds__(256) void gcn_scatter(
    const int* __restrict__ adj, const float* __restrict__ vals,
    const float* __restrict__ sup, float* __restrict__ out) {
  long wid = (long)blockIdx.x * 8 + (threadIdx.x >> 5);
  int lane = threadIdx.x & 31;
  if (wid >= (long)S_T * E_E) return;
  int s = (int)(wid / E_E);
  int e = (int)(wid - (long)s * E_E);
  int dst = adj[s * 2 * E_E + e];
  int src = adj[s * 2 * E_E + E_E + e];
  float v = vals[s * E_E + e];
  const float4 m4 = *(const float4*)(sup + (s * N_N + src) * D_D + lane * 4);
  float* op = out + (s * N_N + dst) * D_D + lane * 4;
  atom_add_f32(op + 0, v * m4.x);
  atom_add_f32(op + 1, v * m4.y);
  atom_add_f32(op + 2, v * m4.z);
  atom_add_f32(op + 3, v * m4.w);
}

// ---------------- bias + relu, in place on d_out ---------------------------
__global__ void bias_relu(float* __restrict__ out, const float* __restrict__ bias) {
  int i = blockIdx.x * blockDim.x + threadIdx.x;
  int stride = gridDim.x * blockDim.x;
  const int n4 = ROWS * D_D / 4;
  const float4* b4 = (const float4*)bias;
  float4* o4 = (float4*)out;
  for (; i < n4; i += stride) {
    float4 x = o4[i];
    float4 bb = b4[i & 31];        // (i*4) % 128 -> float4 index i % 32
    x.x = fmaxf(x.x + bb.x, 0.0f);
    x.y = fmaxf(x.y + bb.y, 0.0f);
    x.z = fmaxf(x.z + bb.z, 0.0f);
    x.w = fmaxf(x.w + bb.w, 0.0f);
    o4[i] = x;
  }
}

// ---------------- sequential LSTM, in place on d_out -----------------------
// Single 1024-thread workgroup (32 wave32). Per step:
//   gates[64,512] = [x_t || h][64,256] @ [w_ih^T ; w_hh^T][256,512] + biases
// A = [x||h] staged k-major in LDS (xh[k][m], exactly 64 KB). Wave w =
// (mt=w>>3, ntb=w&7) owns the four 16x16 gate tiles {i,f,g,o} for batch rows
// mt*16.. and hidden cols ntb*16.. -> cell state c lives in that wave's regs.
__global__ __launch_bounds__(1024) void lstm_seq(
    float* __restrict__ io,                         // [T,64,128] x_t in / h_t out
    const float* __restrict__ w_ih, const float* __restrict__ w_hh,
    const float* __restrict__ b_ih, const float* __restrict__ b_hh,
    const float* __restrict__ h0,  const float* __restrict__ c0) {
  __shared__ float xh[256][64];    // k-major: rows 0..127 = x, 128..255 = h
  const int tid  = threadIdx.x;
  const int lane = tid & 31;
  const int wv   = tid >> 5;
  const int hf   = lane >> 4;
  const int l16  = lane & 15;
  const int mt   = wv >> 3;        // batch-tile 0..3
  const int ntb  = wv & 7;         // hidden-tile 0..7
  const int m0   = mt * 16;
  const int kcol = ntb * 16 + l16; // hidden index this lane's tiles cover

  // warm the 512 KB of weights into L2 (read every one of 3072 steps)
  for (int off = tid * 32; off < 512 * D_D; off += 1024 * 32) {
    __builtin_prefetch(&w_ih[off], 0, 0);
    __builtin_prefetch(&w_hh[off], 0, 0);
  }

  // init h0 into LDS (m fast -> conflict-free LDS stores)
  for (int i = tid; i < B_B * D_D; i += 1024) {
    int m = i & 63, k = i >> 6;
    xh[128 + k][m] = h0[m * D_D + k];
  }
  // init cell-state fragment (C/D layout: reg r -> M = r + 8*half)
  v8f c;
  #pragma unroll
  for (int r = 0; r < 8; ++r) c[r] = c0[(m0 + r + 8 * hf) * D_D + kcol];
  // per-lane gate biases (constant across the 8 C regs: N depends on lane only)
  float biasv[4];
  #pragma unroll
  for (int g = 0; g < 4; ++g) {
    int ng = g * D_D + kcol;
    biasv[g] = b_ih[ng] + b_hh[ng];
  }

  for (int t = 0; t < T_T; ++t) {
    // stage x_t (L2-hot) into LDS, k-major
    const float* xp = io + t * (B_B * D_D);
    for (int i = tid; i < B_B * D_D; i += 1024) {
      int m = i & 63, k = i >> 6;
      xh[k][m] = xp[m * D_D + k];
    }
    __syncthreads();               // x staged + prev-step h writes visible

    v8f acc[4];
    #pragma unroll
    for (int g = 0; g < 4; ++g)
      #pragma unroll
      for (int r = 0; r < 8; ++r) acc[g][r] = biasv[g];

    for (int kk = 0; kk < 64; ++kk) {      // K = 256 (x:128 then h:128)
      int kidx = kk * 4 + hf * 2;
      v2f a;                               // shared across the 4 gate tiles
      a.x = xh[kidx][m0 + l16];
      a.y = xh[kidx + 1][m0 + l16];
      const float* wp = (kidx < D_D) ? (w_ih + kidx) : (w_hh + (kidx - D_D));
      #pragma unroll
      for (int g = 0; g < 4; ++g) {
        int ng = g * D_D + kcol;           // gate row in [0,512)
        v2f b = *(const v2f*)(wp + ng * D_D);   // contiguous k-pair, 8B aligned
        acc[g] = __builtin_amdgcn_wmma_f32_16x16x4_f32(
            false, a, false, b, (short)0, acc[g], false, false);
      }
    }
    __syncthreads();               // everyone done reading h before overwrite

    float* op = io + t * (B_B * D_D);
    #pragma unroll
    for (int r = 0; r < 8; ++r) {
      float ig = sigmoidf_(acc[0][r]);
      float fg = sigmoidf_(acc[1][r]);
      float gg = tanhf(acc[2][r]);
      float og = sigmoidf_(acc[3][r]);
      float cn = fg * c[r] + ig * gg;
      c[r] = cn;
      float h = og * tanhf(cn);
      int m = m0 + r + 8 * hf;
      xh[128 + kcol][m] = h;       // h for next step
      op[m * D_D + kcol] = h;      // hs[t] output (in place over consumed x_t)
    }
    // next iter's x writes touch rows [0,128) only; the post-x-load barrier
    // orders this iter's h writes against next iter's reads.
  }
}

// ---------------- host-side launch ----------------------------------------
extern "C" void kernel_launch(void* const* d_in, const int* in_sizes, int n_in,
                              void* d_out, int out_size, void* d_ws, size_t ws_size,
                              hipStream_t stream) {
  const int*   adj  = (const int*)  d_in[0];   // [S,2,E]
  const float* vals = (const float*)d_in[1];   // [S,E]
  const float* xs   = (const float*)d_in[2];   // [S,N,128]
  const float* gw   = (const float*)d_in[3];   // [128,128]
  const float* gb   = (const float*)d_in[4];   // [128]
  const float* wih  = (const float*)d_in[5];   // [512,128]
  const float* whh  = (const float*)d_in[6];   // [512,128]
  const float* bih  = (const float*)d_in[7];   // [512]
  const float* bhh  = (const float*)d_in[8];   // [512]
  const float* h0   = (const float*)d_in[9];   // [64,128]
  const float* c0   = (const float*)d_in[10];  // [64,128]
  float* out = (float*)d_out;                  // [3072,64,128]
  float* sup = (float*)d_ws;                   // needs ROWS*128*4 = 100.7 MB

  (void)in_sizes; (void)n_in; (void)out_size; (void)ws_size;

  // 1) zero the accumulation target (d_out doubles as GCN output buffer)
  hipLaunchKernelGGL(zero_f4, dim3(4096), dim3(256), 0, stream,
                     (float4*)out, ROWS * D_D / 4);
  // 2) support = xs @ W   (WMMA fp32, TDM + async-to-LDS staging)
  hipLaunchKernelGGL(support_gemm, dim3(ROWS / 16), dim3(256), 0, stream,
                     xs, gw, sup);
  // 3) scatter edges into d_out (native f32 atomics)
  hipLaunchKernelGGL(gcn_scatter, dim3(S_T * (E_E / 8)), dim3(256), 0, stream,
                     adj, vals, sup, out);
  // 4) relu(out + bias) in place -> this is `cur`/`seq`
  hipLaunchKernelGGL(bias_relu, dim3(4096), dim3(256), 0, stream, out, gb);
  // 5) sequential LSTM in place: d_out becomes hs
  hipLaunchKernelGGL(lstm_seq, dim3(1), dim3(1024), 0, stream,
                     out, wih, whh, bih, bhh, h0, c0);
}